// TopKPathMoE_64965675319851
// MI455X (gfx1250) — compile-verified
//
#include <hip/hip_runtime.h>
#include <math.h>

typedef __attribute__((ext_vector_type(2))) float v2f;
typedef __attribute__((ext_vector_type(8))) float v8f;

#define B_   256
#define G_   4000
#define P_   128
#define D_   12000   // 3*G_
#define H_   64
#define O_   16
#define GH_  128
#define CH_  8
#define C_   5
#define K_   3
#define EPS_ 1e-5f

// ---------------------------------------------------------------------------
// Dense GEMM C = act(A[M,Kd] @ W[Kd,N] + bias[N]) using V_WMMA_F32_16X16X4_F32.
// One wave (32 threads) per 16x16 output tile.
// A-layout: lane m = lane%16 row; VGPR0/1 = K (koff, koff+1), koff = (lane/16)*2
// B-layout: lane n = lane%16 col; VGPR0/1 = rows (koff, koff+1)
// C-layout: lane col n = lane%16; VGPR r -> row r + 8*(lane/16)
// ---------------------------------------------------------------------------
__global__ void gate_gemm_wmma(const float* __restrict__ A,
                               const float* __restrict__ W,
                               const float* __restrict__ bias,
                               float* __restrict__ Cout,
                               int M, int N, int Kd, int do_relu) {
    const int ntiles = N >> 4;
    const int tile   = blockIdx.x;
    const int tm     = tile / ntiles;
    const int tn     = tile % ntiles;
    const int lane   = threadIdx.x;        // 0..31, one full wave
    const int half   = lane >> 4;          // 0 or 1
    const int l16    = lane & 15;
    const int m      = tm * 16 + l16;      // A row this lane feeds
    const int n      = tn * 16 + l16;      // B col this lane feeds
    const int koff   = half * 2;

    v8f acc = {};
    for (int k = 0; k < Kd; k += 4) {
        v2f a, b;
        const float* ap = A + (size_t)m * Kd + (k + koff);
        a.x = ap[0];
        a.y = ap[1];
        const float* bp = W + (size_t)(k + koff) * N + n;
        b.x = bp[0];
        b.y = bp[N];
        acc = __builtin_amdgcn_wmma_f32_16x16x4_f32(
            /*neg_a=*/false, a, /*neg_b=*/false, b,
            /*c_mod=*/(short)0, acc, /*reuse_a=*/false, /*reuse_b=*/false);
    }

    const float bv = bias[n];
    const int row_base = tm * 16 + half * 8;
    #pragma unroll
    for (int r = 0; r < 8; ++r) {
        float v = acc[r] + bv;
        if (do_relu) v = fmaxf(v, 0.0f);
        Cout[(size_t)(row_base + r) * N + n] = v;
    }
}

// ---------------------------------------------------------------------------
// Top-3 per row (lax.top_k tie-break: strict >, ascending scan), sigmoid,
// scatter into dense gate_weights output, emit routing tables.
// ---------------------------------------------------------------------------
__global__ void topk_kernel(const float* __restrict__ logits,
                            float* __restrict__ gate_w_out,   // [B_,P_] in d_out
                            int* __restrict__ tidx,           // [B_,K_]
                            float* __restrict__ tw) {         // [B_,K_]
    const int b = blockIdx.x * blockDim.x + threadIdx.x;
    if (b >= B_) return;
    const float* row = logits + (size_t)b * P_;
    float v0 = -INFINITY, v1 = -INFINITY, v2 = -INFINITY;
    int   i0 = 0, i1 = 0, i2 = 0;
    for (int p = 0; p < P_; ++p) {
        const float v = row[p];
        if (v > v0)      { v2 = v1; i2 = i1; v1 = v0; i1 = i0; v0 = v; i0 = p; }
        else if (v > v1) { v2 = v1; i2 = i1; v1 = v;  i1 = p; }
        else if (v > v2) { v2 = v;  i2 = p; }
    }
    const float w0 = 1.0f / (1.0f + expf(-v0));
    const float w1 = 1.0f / (1.0f + expf(-v1));
    const float w2 = 1.0f / (1.0f + expf(-v2));
    float* gw = gate_w_out + (size_t)b * P_;
    for (int p = 0; p < P_; ++p) gw[p] = 0.0f;
    gw[i0] = w0; gw[i1] = w1; gw[i2] = w2;
    tidx[b * K_ + 0] = i0; tw[b * K_ + 0] = w0;
    tidx[b * K_ + 1] = i1; tw[b * K_ + 1] = w1;
    tidx[b * K_ + 2] = i2; tw[b * K_ + 2] = w2;
}

// ---------------------------------------------------------------------------
// Per-pathway compact gene list (ascending, deterministic). One 128-thread
// block per pathway; chunked count -> LDS prefix sum -> write.
// ---------------------------------------------------------------------------
__global__ void build_glist_kernel(const float* __restrict__ gmask,  // [G_,P_]
                                   int* __restrict__ glist,          // [P_,G_]
                                   int* __restrict__ gcount) {       // [P_]
    const int p = blockIdx.x;
    const int t = threadIdx.x;           // 0..127
    __shared__ int cnts[128];
    __shared__ int offs[128];
    const int g0 = t * 32;
    const int g1 = (g0 + 32 < G_) ? (g0 + 32) : G_;
    int c = 0;
    for (int g = g0; g < g1; ++g)
        if (gmask[(size_t)g * P_ + p] != 0.0f) ++c;
    cnts[t] = c;
    __syncthreads();
    if (t == 0) {
        int s = 0;
        for (int i = 0; i < 128; ++i) { offs[i] = s; s += cnts[i]; }
        gcount[p] = s;
    }
    __syncthreads();
    int o = offs[t];
    int* gl = glist + (size_t)p * G_;
    for (int g = g0; g < g1; ++g)
        if (gmask[(size_t)g * P_ + p] != 0.0f) gl[o++] = g;
}

// ---------------------------------------------------------------------------
// One 64-thread block per (sample b, slot k). Thread j owns hidden unit j.
// Only masked genes of W1 are touched (~10% of its 393 MB). x values staged
// through LDS so the W1 reads (coalesced across j) dominate.
// ---------------------------------------------------------------------------
__global__ void expert_kernel(const float* __restrict__ xr,
                              const float* __restrict__ xc,
                              const float* __restrict__ xm,
                              const int*   __restrict__ glist,
                              const int*   __restrict__ gcount,
                              const int*   __restrict__ tidx,
                              const float* __restrict__ tw,
                              const float* __restrict__ W1,   // [P_,D_,H_]
                              const float* __restrict__ b1,   // [P_,H_]
                              const float* __restrict__ gam,
                              const float* __restrict__ bet,
                              const float* __restrict__ mean,
                              const float* __restrict__ var,
                              const float* __restrict__ W2,   // [P_,H_,O_]
                              const float* __restrict__ b2,   // [P_,O_]
                              float* __restrict__ feat_part) {// [B_,K_,O_]
    const int bk = blockIdx.x;           // 0..B_*K_-1
    const int b  = bk / K_;
    const int j  = threadIdx.x;          // 0..63 (hidden unit)
    const int p  = tidx[bk];
    const float w = tw[bk];
    const int cnt = gcount[p];

    __shared__ int   sg[64];
    __shared__ float sx0[64], sx1[64], sx2[64];
    __shared__ float hlds[64];

    const float* W1p = W1 + (size_t)p * D_ * H_;
    const int* gl = glist + (size_t)p * G_;
    float h = 0.0f;

    for (int base = 0; base < cnt; base += 64) {
        const int nn = ((cnt - base) < 64) ? (cnt - base) : 64;
        if (j < nn) {
            const int g = gl[base + j];
            sg[j]  = g;
            sx0[j] = xr[(size_t)b * G_ + g];
            sx1[j] = xc[(size_t)b * G_ + g];
            sx2[j] = xm[(size_t)b * G_ + g];
        }
        __syncthreads();
        for (int i = 0; i < nn; ++i) {
            const size_t g = (size_t)sg[i];
            h = fmaf(sx0[i], W1p[g * H_ + j], h);
            h = fmaf(sx1[i], W1p[((size_t)G_ + g) * H_ + j], h);
            h = fmaf(sx2[i], W1p[((size_t)2 * G_ + g) * H_ + j], h);
        }
        __syncthreads();
    }

    // bias + BatchNorm(eval) + ReLU
    const int pj = p * H_ + j;
    h += b1[pj];
    h = (h - mean[pj]) * (gam[pj] * rsqrtf(var[pj] + EPS_)) + bet[pj];
    h = fmaxf(h, 0.0f);
    hlds[j] = h;
    __syncthreads();

    if (j < O_) {
        const float* W2p = W2 + (size_t)p * H_ * O_;
        float acc = 0.0f;
        #pragma unroll
        for (int i = 0; i < H_; ++i) acc = fmaf(hlds[i], W2p[i * O_ + j], acc);
        acc += b2[p * O_ + j];
        feat_part[(size_t)bk * O_ + j] = acc * w;
    }
}

// ---------------------------------------------------------------------------
// feat = sum_k feat_part  (fixed order -> deterministic), then tiny classifier.
// ---------------------------------------------------------------------------
__global__ void final_kernel(const float* __restrict__ feat_part,
                             const float* __restrict__ cw1,  // [O_,CH_]
                             const float* __restrict__ cb1,  // [CH_]
                             const float* __restrict__ cw2,  // [CH_,C_]
                             const float* __restrict__ cb2,  // [C_]
                             float* __restrict__ out_logits) {
    const int b = blockIdx.x * blockDim.x + threadIdx.x;
    if (b >= B_) return;
    float f[O_];
    #pragma unroll
    for (int o = 0; o < O_; ++o)
        f[o] = feat_part[((size_t)b * K_ + 0) * O_ + o]
             + feat_part[((size_t)b * K_ + 1) * O_ + o]
             + feat_part[((size_t)b * K_ + 2) * O_ + o];
    float hc[CH_];
    #pragma unroll
    for (int c = 0; c < CH_; ++c) {
        float a = cb1[c];
        #pragma unroll
        for (int o = 0; o < O_; ++o) a = fmaf(f[o], cw1[o * CH_ + c], a);
        hc[c] = fmaxf(a, 0.0f);
    }
    #pragma unroll
    for (int c = 0; c < C_; ++c) {
        float a = cb2[c];
        #pragma unroll
        for (int i = 0; i < CH_; ++i) a = fmaf(hc[i], cw2[i * C_ + c], a);
        out_logits[(size_t)b * C_ + c] = a;
    }
}

// ---------------------------------------------------------------------------
extern "C" void kernel_launch(void* const* d_in, const int* in_sizes, int n_in,
                              void* d_out, int out_size, void* d_ws, size_t ws_size,
                              hipStream_t stream) {
    const float* x_rna     = (const float*)d_in[0];
    const float* x_cnv     = (const float*)d_in[1];
    const float* x_met     = (const float*)d_in[2];
    const float* gene_mask = (const float*)d_in[3];
    const float* gate_w1   = (const float*)d_in[4];
    const float* gate_b1   = (const float*)d_in[5];
    const float* gate_w2   = (const float*)d_in[6];
    const float* gate_b2   = (const float*)d_in[7];
    const float* W1        = (const float*)d_in[8];
    const float* b1        = (const float*)d_in[9];
    const float* bn_gamma  = (const float*)d_in[10];
    const float* bn_beta   = (const float*)d_in[11];
    const float* bn_mean   = (const float*)d_in[12];
    const float* bn_var    = (const float*)d_in[13];
    const float* W2        = (const float*)d_in[14];
    const float* b2        = (const float*)d_in[15];
    const float* cls_w1    = (const float*)d_in[16];
    const float* cls_b1    = (const float*)d_in[17];
    const float* cls_w2    = (const float*)d_in[18];
    const float* cls_b2    = (const float*)d_in[19];

    float* out_logits = (float*)d_out;                    // [B_,C_]
    float* out_gatew  = (float*)d_out + (size_t)B_ * C_;  // [B_,P_]

    // workspace carve-up (256B aligned slabs)
    char* ws = (char*)d_ws;
    float* hidden    = (float*)(ws);                      size_t off = (size_t)B_ * GH_ * 4;
    off = (off + 255) & ~(size_t)255;
    float* glogits   = (float*)(ws + off);                off += (size_t)B_ * P_ * 4;
    off = (off + 255) & ~(size_t)255;
    int*   tidx      = (int*)(ws + off);                  off += (size_t)B_ * K_ * 4;
    off = (off + 255) & ~(size_t)255;
    float* tw        = (float*)(ws + off);                off += (size_t)B_ * K_ * 4;
    off = (off + 255) & ~(size_t)255;
    int*   gcount    = (int*)(ws + off);                  off += (size_t)P_ * 4;
    off = (off + 255) & ~(size_t)255;
    int*   glist     = (int*)(ws + off);                  off += (size_t)P_ * G_ * 4;
    off = (off + 255) & ~(size_t)255;
    float* feat_part = (float*)(ws + off);                off += (size_t)B_ * K_ * O_ * 4;
    (void)ws_size; (void)in_sizes; (void)n_in; (void)out_size;

    // gene-list compaction is independent of the gate path
    build_glist_kernel<<<P_, 128, 0, stream>>>(gene_mask, glist, gcount);

    // gate MLP: [256,4000]@[4000,128] (+bias, relu) then [256,128]@[128,128] (+bias)
    gate_gemm_wmma<<<(B_ / 16) * (GH_ / 16), 32, 0, stream>>>(
        x_rna, gate_w1, gate_b1, hidden, B_, GH_, G_, /*relu=*/1);
    gate_gemm_wmma<<<(B_ / 16) * (P_ / 16), 32, 0, stream>>>(
        hidden, gate_w2, gate_b2, glogits, B_, P_, GH_, /*relu=*/0);

    // top-3 + sigmoid + dense gate_weights output + routing tables
    topk_kernel<<<1, B_, 0, stream>>>(glogits, out_gatew, tidx, tw);

    // sparse expert evaluation: one block per (sample, top-k slot)
    expert_kernel<<<B_ * K_, H_, 0, stream>>>(
        x_rna, x_cnv, x_met, glist, gcount, tidx, tw,
        W1, b1, bn_gamma, bn_beta, bn_mean, bn_var, W2, b2, feat_part);

    // combine + classifier
    final_kernel<<<1, B_, 0, stream>>>(feat_part, cls_w1, cls_b1, cls_w2, cls_b2,
                                       out_logits);
}